// GraphMatLayerFastSCM_66468913873405
// MI455X (gfx1250) — compile-verified
//
#include <hip/hip_runtime.h>

typedef __attribute__((ext_vector_type(2))) float v2f;
typedef __attribute__((ext_vector_type(4))) float v4f;
typedef __attribute__((ext_vector_type(8))) float v8f;
typedef int v4i __attribute__((vector_size(4 * sizeof(int))));

#define GLOBAL_AS __attribute__((address_space(1)))
#define LDS_AS    __attribute__((address_space(3)))

#define NN  128   // N == C == P == 128
#define LDA 132   // padded LDS row stride (floats): +4 breaks bank conflicts

#if __has_builtin(__builtin_amdgcn_global_load_async_to_lds_b128) && \
    __has_builtin(__builtin_amdgcn_s_wait_asynccnt)
#define USE_ASYNC 1
#else
#define USE_ASYNC 0
#endif

// Cooperative 128x128 f32 tile load: global (row stride 128) -> LDS (row stride LDA).
// Async-to-LDS (ASYNCcnt) when the toolchain exposes it; plain b128 loads otherwise.
__device__ __forceinline__ void load_tile(const float* __restrict__ src,
                                          float* dst_lds, int tid)
{
#pragma unroll
    for (int i = 0; i < 16; ++i) {
        int idx = i * 256 + tid;        // 0..4095 float4 slots
        int row = idx >> 5;             // 32 float4 per row
        int cv  = (idx & 31) << 2;      // float column
#if USE_ASYNC
        __builtin_amdgcn_global_load_async_to_lds_b128(
            (GLOBAL_AS v4i*)(src + row * NN + cv),
            (LDS_AS v4i*)(dst_lds + row * LDA + cv), 0, 0);
#else
        *(v4f*)&dst_lds[row * LDA + cv] = *(const v4f*)&src[row * NN + cv];
#endif
    }
}

__device__ __forceinline__ void wait_async_all()
{
#if USE_ASYNC
    __builtin_amdgcn_s_wait_asynccnt(0);
#endif
}

// One block per (g, b): out[g,b] = relu(G[b,g] @ (x[b] @ W[g]^T + bias[g]))
// LDS = 2 x 66KB buffers -> 132KB -> 2 blocks resident per WGP (320KB LDS),
// so one block's WMMA phase hides the other's load phase.
__global__ __launch_bounds__(256) void gml_fused_kernel(
    const float* __restrict__ G, const float* __restrict__ X,
    const float* __restrict__ W, const float* __restrict__ bias,
    float* __restrict__ out, int Bsz, int GSsz)
{
    __shared__ float xs[NN * LDA];  // x tile; later reused as multi_x^T  [p][n]
    __shared__ float ws[NN * LDA];  // W[g] as [p][c]; later reused for G[b,g] [k][l]

    const int b_idx = blockIdx.x;
    const int g_idx = blockIdx.y;
    const float* Gt = G + ((size_t)b_idx * GSsz + g_idx) * (size_t)(NN * NN);
    const float* xt = X + (size_t)b_idx * (size_t)(NN * NN);
    const float* Wt = W + (size_t)g_idx * (size_t)(NN * NN);
    const float* bt = bias + (size_t)g_idx * NN;
    float* outt = out + ((size_t)g_idx * Bsz + b_idx) * (size_t)(NN * NN);

    const int tid = threadIdx.x;

    // ---- phase 1: stage x and W into LDS; prefetch G so its HBM fetch overlaps stage A ----
    load_tile(xt, xs, tid);
    load_tile(Wt, ws, tid);
    // G tile = 64KB = 512 x 128B cachelines; 2 lines per thread
    __builtin_prefetch(Gt + (size_t)tid * 32, 0, 3);
    __builtin_prefetch(Gt + (size_t)(tid + 256) * 32, 0, 3);
    wait_async_all();
    __syncthreads();

    const int lane  = tid & 31;
    const int wv    = tid >> 5;               // wave id 0..7 -> row-tile
    const int l16   = lane & 15;
    const int lhalf = lane >> 4;              // 0 | 1
    const int koff  = lhalf << 1;             // A/B frag: K = k + koff + {0,1}
    const int arow  = wv * 16 + l16;          // A-matrix row for this lane

    // ---- Stage A: multi_x = x @ W^T + bias  (rows 16*wv .. 16*wv+15) ----
    v8f acc[8];
#pragma unroll
    for (int tn = 0; tn < 8; ++tn) {
        float bv = bt[tn * 16 + l16];         // bias broadcast per output column
        v8f t = {bv, bv, bv, bv, bv, bv, bv, bv};
        acc[tn] = t;
    }
    for (int k = 0; k < NN; k += 4) {
        v2f a = *(const v2f*)&xs[arow * LDA + k + koff];
#pragma unroll
        for (int tn = 0; tn < 8; ++tn) {
            v2f bb = *(const v2f*)&ws[(tn * 16 + l16) * LDA + k + koff];
            acc[tn] = __builtin_amdgcn_wmma_f32_16x16x4_f32(
                false, a, false, bb, (short)0, acc[tn], false, false);
        }
    }
    __syncthreads();  // all reads of xs and ws done

    // ---- phase 3: G -> ws (async) concurrent with multi_x^T writeback -> xs ----
    load_tile(Gt, ws, tid);
    {
        const int nbase = wv * 16 + (lhalf << 3);   // C/D frag: M = nbase + r
#pragma unroll
        for (int tn = 0; tn < 8; ++tn) {
            int p = tn * 16 + l16;                  // output column -> msT row
            v4f lo = __builtin_shufflevector(acc[tn], acc[tn], 0, 1, 2, 3);
            v4f hi = __builtin_shufflevector(acc[tn], acc[tn], 4, 5, 6, 7);
            *(v4f*)&xs[p * LDA + nbase]     = lo;
            *(v4f*)&xs[p * LDA + nbase + 4] = hi;
        }
    }
    wait_async_all();
    __syncthreads();

    // ---- Stage B: out = relu(G @ multi_x) ----
    v8f oacc[8];
#pragma unroll
    for (int tn = 0; tn < 8; ++tn) {
        v8f z = {0.f, 0.f, 0.f, 0.f, 0.f, 0.f, 0.f, 0.f};
        oacc[tn] = z;
    }
    for (int k = 0; k < NN; k += 4) {
        v2f a = *(const v2f*)&ws[arow * LDA + k + koff];   // G[m][l] fragment
#pragma unroll
        for (int tn = 0; tn < 8; ++tn) {
            // B frag: element (K=l, N=p) = multi_x[l][p] = msT[p][l] -> contiguous float2
            v2f bb = *(const v2f*)&xs[(tn * 16 + l16) * LDA + k + koff];
            oacc[tn] = __builtin_amdgcn_wmma_f32_16x16x4_f32(
                false, a, false, bb, (short)0, oacc[tn], false, false);
        }
    }

    // ---- relu + coalesced store ----
    const int mbase = wv * 16 + (lhalf << 3);
#pragma unroll
    for (int tn = 0; tn < 8; ++tn) {
        int col = tn * 16 + l16;
#pragma unroll
        for (int r = 0; r < 8; ++r) {
            outt[(size_t)(mbase + r) * NN + col] = fmaxf(oacc[tn][r], 0.0f);
        }
    }
}

extern "C" void kernel_launch(void* const* d_in, const int* in_sizes, int n_in,
                              void* d_out, int out_size, void* d_ws, size_t ws_size,
                              hipStream_t stream) {
    // setup_inputs order: G [B,GS,N,N], x [B,N,C], W [GS,P,C], b [GS,P] (all f32)
    const float* G    = (const float*)d_in[0];
    const float* x    = (const float*)d_in[1];
    const float* W    = (const float*)d_in[2];
    const float* bias = (const float*)d_in[3];
    float* out = (float*)d_out;

    const int Bsz  = 256;
    const int GSsz = 4;
    dim3 grid(Bsz, GSsz);
    gml_fused_kernel<<<grid, 256, 0, stream>>>(G, x, W, bias, out, Bsz, GSsz);
}